// LocalCosineSimilarity_37383395344620
// MI455X (gfx1250) — compile-verified
//
#include <hip/hip_runtime.h>

typedef __attribute__((ext_vector_type(2))) float v2f;
typedef __attribute__((ext_vector_type(4))) float v4f;
typedef __attribute__((ext_vector_type(8))) float v8f;

#define NEGV  (-1e9f)
#define BB_   8
#define N1_   1023
#define NN_   1024
#define DD_   768
#define FF_   8
#define NTOT_ 8192

// One wave handles one (b, f, 16-row tile):
//  - WMMA f32 16x16x4 accumulation over K=768 for A.B^T, A.A^T, B.B^T
//  - diagonals -> 16 cosine similarities
//  - NEG-fill the 16x1024 output tile (NT stores), plug diagonal values.
__global__ __launch_bounds__(256) void lcs_wmma_kernel(
    const float* __restrict__ lhs,   // [8, 1023, 768]
    const float* __restrict__ rhs,   // [8, 8192, 768]
    float* __restrict__ out)         // [8, 1023, 8192]
{
  __shared__ float shm[8][3][16];    // [wave][AB,AA,BB][diag 16]

  const int lane = threadIdx.x & 31;
  const int wv   = threadIdx.x >> 5;
  const int g    = blockIdx.x * 8 + wv;      // 0..4095
  const int t    = g & 63;                   // 16-row tile index (0..63)
  const int f    = (g >> 6) & 7;             // block column (0..7)
  const int b    = g >> 9;                   // batch (0..7)

  const int n0   = t * 16;
  const int rl   = lane & 15;
  const int nrow = n0 + rl;                          // lhs row this lane owns
  const int nld  = nrow > (N1_ - 1) ? (N1_ - 1) : nrow;
  const int mrow = f * NN_ + 1 + nrow;               // matching rhs row (diag col)
  const int mld  = mrow > (NTOT_ - 1) ? (NTOT_ - 1) : mrow;
  const int koff = (lane >> 4) * 2;                  // lanes 0-15: k+0..1, 16-31: k+2..3

  const float* ap = lhs + ((size_t)b * N1_   + nld) * DD_ + koff;
  const float* bp = rhs + ((size_t)b * NTOT_ + mld) * DD_ + koff;

  v8f accAB = {}; v8f accAA = {}; v8f accBB = {};
  // A-operand (16x4) and B-operand (4x16, B = rows^T) have identical per-lane
  // layouts for row-major source rows, so one register pair serves both roles.
#pragma unroll 4
  for (int k = 0; k < DD_; k += 4) {
    v2f av = *(const v2f*)(ap + k);
    v2f bv = *(const v2f*)(bp + k);
    accAB = __builtin_amdgcn_wmma_f32_16x16x4_f32(false, av, false, bv,
                                                  (short)0, accAB, false, false);
    accAA = __builtin_amdgcn_wmma_f32_16x16x4_f32(false, av, false, av,
                                                  (short)0, accAA, false, false);
    accBB = __builtin_amdgcn_wmma_f32_16x16x4_f32(false, bv, false, bv,
                                                  (short)0, accBB, false, false);
  }

  // Extract diagonals: C layout -> diag(i) i<8 at lane i comp i,
  // diag(i) i>=8 at lane i+16 comp i-8.  Stage through LDS (same-wave, in-order).
#pragma unroll
  for (int r = 0; r < 8; ++r) {
    if (lane == r) {
      shm[wv][0][r] = accAB[r];
      shm[wv][1][r] = accAA[r];
      shm[wv][2][r] = accBB[r];
    }
    if (lane == r + 24) {
      shm[wv][0][r + 8] = accAB[r];
      shm[wv][1][r + 8] = accAA[r];
      shm[wv][2][r + 8] = accBB[r];
    }
  }
  __syncthreads();

  float res = 0.0f;
  if (lane < 16) {
    float dab = shm[wv][0][lane];
    float daa = shm[wv][1][lane];
    float dbb = shm[wv][2][lane];
    res = dab * rsqrtf(daa * dbb);   // cosine; EXP==1 so no pow needed
  }

  // NEG-fill this wave's 16x1024 tile with non-temporal B128 stores.
  float* base = out + (size_t)b * N1_ * NTOT_ + (size_t)f * NN_;
  const v4f negv = {NEGV, NEGV, NEGV, NEGV};
#pragma unroll 1
  for (int i = 0; i < 16; ++i) {
    int n = n0 + i;
    if (n >= N1_) break;             // wave-uniform
    float* rowp = base + (size_t)n * NTOT_;
#pragma unroll
    for (int c = 0; c < 8; ++c)
      __builtin_nontemporal_store(negv, (v4f*)(rowp + c * 128 + lane * 4));
  }

  // Plug diagonal values (same-wave stores are ordered after the fill).
  if (lane < 16 && nrow < N1_)
    base[(size_t)nrow * NTOT_ + (nrow + 1)] = res;
}

extern "C" void kernel_launch(void* const* d_in, const int* in_sizes, int n_in,
                              void* d_out, int out_size, void* d_ws, size_t ws_size,
                              hipStream_t stream) {
  const float* lhs = (const float*)d_in[0];   // 8*1023*768
  const float* rhs = (const float*)d_in[1];   // 8*8192*768
  float* out = (float*)d_out;                 // 8*1023*8192
  (void)in_sizes; (void)n_in; (void)out_size; (void)d_ws; (void)ws_size;

  // 4096 waves = 512 blocks x 8 waves; one wave per (b, f, 16-row tile).
  lcs_wmma_kernel<<<512, 256, 0, stream>>>(lhs, rhs, out);
}